// SoftMLLoss_39814346834536
// MI455X (gfx1250) — compile-verified
//
#include <hip/hip_runtime.h>

// ---------------------------------------------------------------------------
// SoftML margin loss, MI455X (gfx1250, wave32).
// Memory-bound streaming pass (167.8 MB @ 23.3 TB/s ~ 7.2 us floor):
//   - 4 samples per lane per iter: 3x b128 loads for distances, 2x b128 for labels
//   - f32 stable softplus in registers
//   - wave 32->1 reduction via V_WMMA_F32_16X16X4_F32 (exact f32)
//   - deterministic LDS block combine + second-pass finalize (no float atomics)
// ---------------------------------------------------------------------------

#define LAMBDA_MARGIN 500.0f

typedef __attribute__((ext_vector_type(2))) float v2f;
typedef __attribute__((ext_vector_type(8))) float v8f;

__device__ __forceinline__ float stable_softplus(float x) {
    // log1p(exp(x)) computed stably; exp(-|x|) underflows harmlessly to 0.
    return fmaxf(x, 0.0f) + log1pf(expf(-fabsf(x)));
}

__device__ __forceinline__ float per_sample_loss(float d0, float d1, float d2,
                                                 int dl, int rl) {
    float sp0 = stable_softplus(LAMBDA_MARGIN - d0);
    float sp1 = stable_softplus(LAMBDA_MARGIN - d1);
    float sp2 = stable_softplus(LAMBDA_MARGIN - d2);
    float c0 = d0 + sp1 + sp2;                       // dl==0: d0 + (sum - sp0)
    float c1 = d1 + sp0 + sp2;                       // dl==1: d1 + (sum - sp1)
    float c2 = (rl == 0) ? sp1 : ((rl == 1) ? sp0 : 0.0f);   // dl==2 (reject)
    return (dl == 0) ? c0 : ((dl == 1) ? c1 : ((dl == 2) ? c2 : 0.0f));
}

// Exact f32 sum of the 32 per-lane values in a wave, using one
// v_wmma_f32_16x16x4_f32. A: lane l<16 -> A[l][0], lane l+16 -> A[l][2]
// (other K slots zero); B = ones. => D[m,n] = p[m] + p[m+16] = r[m].
// C-layout: lane l's 8 acc VGPRs hold r[j + (l<16?0:8)], j=0..7; summing
// them gives the half-sums; shfl_xor(16) folds the halves.
// EXEC must be all ones here: callers invoke this with no divergence.
__device__ __forceinline__ float wave_reduce_wmma(float p) {
    v2f a; a.x = p;    a.y = 0.0f;
    v2f b; b.x = 1.0f; b.y = 1.0f;
    v8f c = {};
    c = __builtin_amdgcn_wmma_f32_16x16x4_f32(
        /*neg_a=*/false, a, /*neg_b=*/false, b,
        /*c_mod=*/(short)0, c, /*reuse_a=*/false, /*reuse_b=*/false);
    float s = ((c[0] + c[1]) + (c[2] + c[3])) + ((c[4] + c[5]) + (c[6] + c[7]));
    return s + __shfl_xor(s, 16, 32);   // total, replicated in every lane
}

__device__ __forceinline__ void block_store_sum(float waveSum, float* dst,
                                                float scale) {
    __shared__ float warpSums[8];
    int wave = threadIdx.x >> 5;
    if ((threadIdx.x & 31) == 0) warpSums[wave] = waveSum;
    __syncthreads();
    if (threadIdx.x == 0) {
        float s = 0.0f;
        int nw = blockDim.x >> 5;
        for (int w = 0; w < nw; ++w) s += warpSums[w];   // fixed order: deterministic
        *dst = s * scale;
    }
}

__global__ void __launch_bounds__(256)
softml_partial_kernel(const float* __restrict__ dist,
                      const int* __restrict__ dlab,
                      const int* __restrict__ rlab,
                      float* __restrict__ partials,
                      int batch) {
    const int ntiles   = batch >> 2;                       // 4 samples / tile
    const int nthreads = gridDim.x * blockDim.x;
    const int tid      = blockIdx.x * blockDim.x + threadIdx.x;

    const float4* __restrict__ dist4 = (const float4*)dist; // 12 floats = 3 x float4
    const int4*   __restrict__ dl4   = (const int4*)dlab;
    const int4*   __restrict__ rl4   = (const int4*)rlab;

    float acc = 0.0f;
    for (int t = tid; t < ntiles; t += nthreads) {
        float4 f0 = dist4[3 * t + 0];
        float4 f1 = dist4[3 * t + 1];
        float4 f2 = dist4[3 * t + 2];
        int4   a  = dl4[t];
        int4   b  = rl4[t];
        acc += per_sample_loss(f0.x, f0.y, f0.z, a.x, b.x);
        acc += per_sample_loss(f0.w, f1.x, f1.y, a.y, b.y);
        acc += per_sample_loss(f1.z, f1.w, f2.x, a.z, b.z);
        acc += per_sample_loss(f2.y, f2.z, f2.w, a.w, b.w);
    }
    // Tail (batch not multiple of 4) — negligible, single thread, still
    // reaches the uniform reduction below with all lanes active.
    if (tid == 0) {
        for (int i = ntiles << 2; i < batch; ++i)
            acc += per_sample_loss(dist[3 * i], dist[3 * i + 1], dist[3 * i + 2],
                                   dlab[i], rlab[i]);
    }

    float waveSum = wave_reduce_wmma(acc);                 // EXEC all ones here
    block_store_sum(waveSum, &partials[blockIdx.x], 1.0f);
}

__global__ void __launch_bounds__(256)
softml_finalize_kernel(const float* __restrict__ partials, int n,
                       float* __restrict__ out, float inv_batch) {
    float acc = 0.0f;
    for (int i = threadIdx.x; i < n; i += blockDim.x)      // fixed stride order
        acc += partials[i];
    float waveSum = wave_reduce_wmma(acc);
    block_store_sum(waveSum, out, inv_batch);
}

extern "C" void kernel_launch(void* const* d_in, const int* in_sizes, int n_in,
                              void* d_out, int out_size, void* d_ws, size_t ws_size,
                              hipStream_t stream) {
    const float* dist = (const float*)d_in[0];   // (B, 3) f32
    const int*   dlab = (const int*)d_in[1];     // (B,)   i32
    const int*   rlab = (const int*)d_in[2];     // (B,)   i32
    float*       out  = (float*)d_out;           // scalar f32

    const int batch = in_sizes[1];               // B = 8,388,608

    const int THREADS = 256;                     // 8 wave32 per block
    const int BLOCKS  = 2048;                    // 8 KB of d_ws for partials
    float* partials = (float*)d_ws;

    softml_partial_kernel<<<BLOCKS, THREADS, 0, stream>>>(dist, dlab, rlab,
                                                          partials, batch);
    softml_finalize_kernel<<<1, THREADS, 0, stream>>>(partials, BLOCKS, out,
                                                      1.0f / (float)batch);
}